// GCN_80616536146118
// MI455X (gfx1250) — compile-verified
//
#include <hip/hip_runtime.h>

typedef float v2f __attribute__((ext_vector_type(2)));
typedef float v8f __attribute__((ext_vector_type(8)));

#define NNODES 100000
#define DFEAT  64
#define NTILES (NNODES / 16)     // 6250 row tiles, exact
#define WSTR   65                // padded LDS stride (bank-conflict free)

// ---------------------------------------------------------------------------
// h[N,64] = x[N,64] @ W[64,64]  via V_WMMA_F32_16X16X4_F32
// One wave computes a 16-row strip across all 64 output columns.
// ---------------------------------------------------------------------------
__global__ __launch_bounds__(256) void gemm64_wmma(const float* __restrict__ x,
                                                   const float* __restrict__ W,
                                                   float* __restrict__ h,
                                                   int nRowTiles) {
    __shared__ float Wl[64 * WSTR];
    // cooperative load of W (64x64) into padded LDS
    for (int i = threadIdx.x; i < 64 * 64; i += 256) {
        int r = i >> 6, c = i & 63;
        Wl[r * WSTR + c] = W[i];
    }
    __syncthreads();

    const int wave = threadIdx.x >> 5;
    const int lane = threadIdx.x & 31;
    const int tile = blockIdx.x * 8 + wave;
    if (tile >= nRowTiles) return;   // wave-uniform: EXEC all-1s inside WMMA

    const int hl  = lane >> 4;       // 0: K even pair, 1: K odd pair (per ISA A layout)
    const int l15 = lane & 15;       // M (for A) / N (for B,C,D)
    const int rowA = tile * 16 + l15;

    // Preload A strip: 16 k-steps of K=4; lane holds {K=2*hl, K=2*hl+1}
    v2f a[16];
    const float* xrow = x + (size_t)rowA * DFEAT + hl * 2;
    #pragma unroll
    for (int k = 0; k < 16; ++k) {
        a[k].x = xrow[k * 4 + 0];
        a[k].y = xrow[k * 4 + 1];
    }

    #pragma unroll
    for (int nt = 0; nt < 4; ++nt) {
        v8f acc = {};
        const int col = nt * 16 + l15;
        #pragma unroll
        for (int k = 0; k < 16; ++k) {
            const int krow = k * 4 + hl * 2;
            v2f b;
            b.x = Wl[(krow + 0) * WSTR + col];
            b.y = Wl[(krow + 1) * WSTR + col];
            acc = __builtin_amdgcn_wmma_f32_16x16x4_f32(
                false, a[k], false, b, (short)0, acc, false, false);
        }
        // D layout: VGPR r -> M = r + 8*hl, N = l15
        #pragma unroll
        for (int r = 0; r < 8; ++r) {
            const int row = tile * 16 + r + 8 * hl;
            h[(size_t)row * DFEAT + col] = acc[r];
        }
    }
}

// ---------------------------------------------------------------------------
// Degree / norm helpers:  dinv[i] = rsqrt(1 + indegree(i))   (self-loop incl.)
// ---------------------------------------------------------------------------
__global__ void deg_init(float* __restrict__ deg, int n) {
    int i = blockIdx.x * 256 + threadIdx.x;
    if (i < n) deg[i] = 1.0f;                      // self loop
}
__global__ void deg_acc(const int* __restrict__ dst, float* __restrict__ deg, int e) {
    int i = blockIdx.x * 256 + threadIdx.x;
    if (i < e) atomicAdd(&deg[dst[i]], 1.0f);
}
__global__ void deg_finish(float* __restrict__ deg, int n) {
    int i = blockIdx.x * 256 + threadIdx.x;
    if (i < n) deg[i] = rsqrtf(deg[i]);            // deg >= 1 always
}

__global__ void zero_f(float* __restrict__ p, int n) {
    int i = blockIdx.x * 256 + threadIdx.x;
    if (i < n) p[i] = 0.0f;
}

// ---------------------------------------------------------------------------
// Edge scatter: agg[dst] += h[src] * dinv[src]*dinv[dst]
// 16 threads per edge, float4 per thread (coalesced gather + L2 atomics).
// ---------------------------------------------------------------------------
__global__ __launch_bounds__(256) void edge_scatter(const float* __restrict__ h,
                                                    const int* __restrict__ src,
                                                    const int* __restrict__ dst,
                                                    const float* __restrict__ dinv,
                                                    float* __restrict__ agg, int E) {
    int tid = blockIdx.x * 256 + threadIdx.x;
    int e = tid >> 4;
    if (e >= E) return;
    int c4 = (tid & 15) * 4;
    int s = src[e], d = dst[e];
    float w = dinv[s] * dinv[d];
    const float4 hv = *(const float4*)(h + (size_t)s * DFEAT + c4);
    float* op = agg + (size_t)d * DFEAT + c4;
    atomicAdd(op + 0, hv.x * w);
    atomicAdd(op + 1, hv.y * w);
    atomicAdd(op + 2, hv.z * w);
    atomicAdd(op + 3, hv.w * w);
}

// ---------------------------------------------------------------------------
// Finalize: out = [relu](agg + h*dinv^2 (self loop) + bias)
// ---------------------------------------------------------------------------
__global__ void finalize(const float* __restrict__ agg, const float* __restrict__ h,
                         const float* __restrict__ dinv, const float* __restrict__ bias,
                         float* __restrict__ out, int do_relu, int n) {
    int i = blockIdx.x * 256 + threadIdx.x;
    if (i >= n) return;
    int node = i >> 6, c = i & 63;
    float di = dinv[node];
    float v = agg[i] + h[i] * (di * di) + bias[c];
    if (do_relu) v = fmaxf(v, 0.0f);
    out[i] = v;
}

// ---------------------------------------------------------------------------
extern "C" void kernel_launch(void* const* d_in, const int* in_sizes, int n_in,
                              void* d_out, int out_size, void* d_ws, size_t ws_size,
                              hipStream_t stream) {
    const float* x0 = (const float*)d_in[0];         // node_embed [N,64]
    const int*   me = (const int*)d_in[1];           // message_edge [2,E]
    const float* W1 = (const float*)d_in[2];
    const float* b1 = (const float*)d_in[3];
    const float* W2 = (const float*)d_in[4];
    const float* b2 = (const float*)d_in[5];
    const float* W3 = (const float*)d_in[6];
    const float* b3 = (const float*)d_in[7];

    const int E = in_sizes[1] / 2;
    const int* src = me;
    const int* dst = me + E;
    const int NF = NNODES * DFEAT;

    // workspace layout (floats)
    float* ws   = (float*)d_ws;
    float* dinv = ws;                                // N
    float* h    = ws + 100352;                       // N*64 (aligned offset)
    float* xb   = h + NF;                            // N*64
    float* agg  = xb + NF;                           // N*64

    float* out = (float*)d_out;

    const int TB = 256;
    const int gN   = (NNODES + TB - 1) / TB;
    const int gE   = (E + TB - 1) / TB;
    const int gE16 = (E * 16 + TB - 1) / TB;
    const int gNF  = (NF + TB - 1) / TB;
    const int gGEMM = (NTILES + 7) / 8;

    // normalization
    deg_init<<<gN, TB, 0, stream>>>(dinv, NNODES);
    deg_acc<<<gE, TB, 0, stream>>>(dst, dinv, E);
    deg_finish<<<gN, TB, 0, stream>>>(dinv, NNODES);

    // ---- layer 1 ----
    gemm64_wmma<<<gGEMM, TB, 0, stream>>>(x0, W1, h, NTILES);
    zero_f<<<gNF, TB, 0, stream>>>(agg, NF);
    edge_scatter<<<gE16, TB, 0, stream>>>(h, src, dst, dinv, agg, E);
    finalize<<<gNF, TB, 0, stream>>>(agg, h, dinv, b1, xb, 1, NF);

    // ---- layer 2 ----
    gemm64_wmma<<<gGEMM, TB, 0, stream>>>(xb, W2, h, NTILES);
    zero_f<<<gNF, TB, 0, stream>>>(agg, NF);
    edge_scatter<<<gE16, TB, 0, stream>>>(h, src, dst, dinv, agg, E);
    finalize<<<gNF, TB, 0, stream>>>(agg, h, dinv, b2, xb, 1, NF);

    // ---- layer 3 (no relu), accumulate directly into d_out ----
    gemm64_wmma<<<gGEMM, TB, 0, stream>>>(xb, W3, h, NTILES);
    zero_f<<<gNF, TB, 0, stream>>>(out, NF);
    edge_scatter<<<gE16, TB, 0, stream>>>(h, src, dst, dinv, out, E);
    finalize<<<gNF, TB, 0, stream>>>(out, h, dinv, b3, out, 0, NF);
}